// GNN_NoTemporal_65163243815592
// MI455X (gfx1250) — compile-verified
//
#include <hip/hip_runtime.h>
#include <hip/hip_bf16.h>

// ---------------------------------------------------------------------------
// GNN (temporal-mean -> 2x GCNConv -> mean-pool -> linear head) for MI455X.
// Dominant costs: x read (38.4 MB), 2x 1M-edge scatter-add (L2-resident).
// Dense transform h1@W2 (64000x64x64) done with V_WMMA_F32_16X16X4_F32,
// full fp32 precision (workload is bandwidth bound; fp32 matrix pipe is free).
// W2 staged in LDS pair-interleaved so each B fragment is one ds_load_b64.
// ---------------------------------------------------------------------------

#define Bv   32
#define Sv   50
#define Nv   2000
#define Fv   3
#define Dv   64
#define Ev   1000000
#define NUM_NODES (Bv * Nv)   // 64000

typedef __attribute__((ext_vector_type(2))) float v2f;
typedef __attribute__((ext_vector_type(8))) float v8f;

// ---- 1. temporal mean: h0[node][f] = mean_s x[b][s][n][f] -------------------
__global__ void k_temporal_mean(const float* __restrict__ x, float* __restrict__ h0) {
    int tid = blockIdx.x * blockDim.x + threadIdx.x;      // over 64000*3
    if (tid >= NUM_NODES * Fv) return;
    int node = tid / Fv;
    int f    = tid % Fv;
    int b = node / Nv;
    int n = node % Nv;
    const float* p = x + ((long long)b * Sv * Nv + n) * Fv + f;
    float s = 0.0f;
    #pragma unroll 5
    for (int t = 0; t < Sv; ++t) s += p[(long long)t * Nv * Fv];
    h0[tid] = s * (1.0f / (float)Sv);
}

// ---- 2a. degree init (self-loop contributes 1) ------------------------------
__global__ void k_deg_init(float* __restrict__ deg) {
    int i = blockIdx.x * blockDim.x + threadIdx.x;
    if (i < NUM_NODES) deg[i] = 1.0f;
}

// ---- 2b. degree scatter over edge destinations ------------------------------
__global__ void k_deg_scatter(const int* __restrict__ ei, float* __restrict__ deg) {
    int e = blockIdx.x * blockDim.x + threadIdx.x;
    if (e >= Ev) return;
    int d = ei[Ev + e];                 // dst row
    atomicAdd(deg + d, 1.0f);
}

// ---- 2c. dinv = rsqrt(deg) in place -----------------------------------------
__global__ void k_dinv(float* __restrict__ deg) {
    int i = blockIdx.x * blockDim.x + threadIdx.x;
    if (i < NUM_NODES) {
        float d = deg[i];
        deg[i] = (d > 0.0f) ? __frsqrt_rn(d) : 0.0f;
    }
}

// ---- 3. per-edge norm = dinv[src]*dinv[dst] ---------------------------------
__global__ void k_norm(const int* __restrict__ ei, const float* __restrict__ dinv,
                       float* __restrict__ norm) {
    int e = blockIdx.x * blockDim.x + threadIdx.x;
    if (e >= Ev) return;
    norm[e] = dinv[ei[e]] * dinv[ei[Ev + e]];
}

// ---- 4. t1 = h0 @ W1 (K=3, pure VALU; memory bound) -------------------------
__global__ void k_gemm1(const float* __restrict__ h0, const float* __restrict__ W1,
                        float* __restrict__ T1) {
    int tid = blockIdx.x * blockDim.x + threadIdx.x;      // over 64000*64
    if (tid >= NUM_NODES * Dv) return;
    int node = tid >> 6;
    int d    = tid & 63;
    const float* h = h0 + node * 3;
    T1[tid] = fmaf(h[0], W1[d], fmaf(h[1], W1[Dv + d], h[2] * W1[2 * Dv + d]));
}

// ---- 5. agg init: bias + self-loop message (norm_ii = dinv^2) ---------------
__global__ void k_agg_init(const float* __restrict__ T, const float* __restrict__ dinv,
                           const float* __restrict__ bias, float* __restrict__ Agg) {
    int tid = blockIdx.x * blockDim.x + threadIdx.x;      // over 64000*64
    if (tid >= NUM_NODES * Dv) return;
    int node = tid >> 6;
    int d    = tid & 63;
    float di = dinv[node];
    Agg[tid] = fmaf(T[tid], di * di, bias[d]);
}

// ---- 6. edge scatter: Agg[dst] += T[src] * norm  (16 threads/edge, float4) --
__global__ void k_edge_scatter(const int* __restrict__ ei, const float* __restrict__ norm,
                               const float* __restrict__ T, float* __restrict__ Agg) {
    int tid = blockIdx.x * blockDim.x + threadIdx.x;      // 16M threads
    int e = tid >> 4;
    if (e >= Ev) return;
    int c = (tid & 15) * 4;
    int s = ei[e];
    int d = ei[Ev + e];
    float w = norm[e];
    const float4 v = *reinterpret_cast<const float4*>(T + s * Dv + c);
    float* out = Agg + d * Dv + c;
    atomicAdd(out + 0, v.x * w);
    atomicAdd(out + 1, v.y * w);
    atomicAdd(out + 2, v.z * w);
    atomicAdd(out + 3, v.w * w);
}

// ---- 7. T2 = relu(Agg1) @ W2 via V_WMMA_F32_16X16X4_F32 ---------------------
// 8 waves/block, each wave: one 16-row strip x all 64 cols (4 accumulators).
// A frag (16x4 f32): lanes 0-15 K={k,k+1}, lanes 16-31 K={k+2,k+3}, M=lane&15.
// B frag (4x16 f32): same K split, N=lane&15. D: vgpr r -> (M=r+8*half, N=lane&15).
// W2 is staged pair-interleaved: WlP[p*64+n] = {W[2p][n], W[2p+1][n]} so each
// half-wave's B fragment is one contiguous ds_load_b64 into an even VGPR pair.
__global__ void __launch_bounds__(256)
k_gemm2_wmma(const float* __restrict__ Hin, const float* __restrict__ W,
             float* __restrict__ Out) {
    __shared__ v2f WlP[(Dv / 2) * Dv];                    // 16 KB
    for (int i = threadIdx.x; i < (Dv / 2) * Dv; i += 256) {
        int p = i >> 6;                                   // K-pair index 0..31
        int n = i & 63;                                   // column
        v2f w;
        w.x = W[(2 * p) * Dv + n];
        w.y = W[(2 * p + 1) * Dv + n];
        WlP[i] = w;
    }
    __syncthreads();

    const int wave = threadIdx.x >> 5;
    const int lane = threadIdx.x & 31;
    const int half = lane >> 4;                            // 0 or 1
    const int l    = lane & 15;
    const int m0   = (blockIdx.x * 8 + wave) * 16;

    v8f acc0 = {}, acc1 = {}, acc2 = {}, acc3 = {};

    #pragma unroll
    for (int k = 0; k < Dv; k += 4) {
        const int ka = k + 2 * half;                       // this half-wave's K pair
        const int pa = (k >> 1) + half;                    // pair index in WlP
        // A fragment with fused ReLU (Hin is pre-activation agg1)
        const v2f av = *reinterpret_cast<const v2f*>(Hin + (m0 + l) * Dv + ka);
        v2f a;
        a.x = fmaxf(av.x, 0.0f);
        a.y = fmaxf(av.y, 0.0f);
        // B fragments for the 4 N-tiles: single b64 LDS load each
        const v2f b0 = WlP[pa * Dv +  0 + l];
        const v2f b1 = WlP[pa * Dv + 16 + l];
        const v2f b2 = WlP[pa * Dv + 32 + l];
        const v2f b3 = WlP[pa * Dv + 48 + l];

        acc0 = __builtin_amdgcn_wmma_f32_16x16x4_f32(false, a, false, b0, (short)0, acc0, false, false);
        acc1 = __builtin_amdgcn_wmma_f32_16x16x4_f32(false, a, false, b1, (short)0, acc1, false, false);
        acc2 = __builtin_amdgcn_wmma_f32_16x16x4_f32(false, a, false, b2, (short)0, acc2, false, false);
        acc3 = __builtin_amdgcn_wmma_f32_16x16x4_f32(false, a, false, b3, (short)0, acc3, false, false);
    }

    const int row0 = m0 + 8 * half;
    #pragma unroll
    for (int r = 0; r < 8; ++r) {
        float* o = Out + (row0 + r) * Dv + l;
        o[0]  = acc0[r];
        o[16] = acc1[r];
        o[32] = acc2[r];
        o[48] = acc3[r];
    }
}

// ---- 8. per-graph mean pool of relu(Agg2) + head ----------------------------
__global__ void __launch_bounds__(256)
k_pool_head(const float* __restrict__ Agg2, const float* __restrict__ Wh,
            const float* __restrict__ bh, float* __restrict__ Out) {
    __shared__ float part[4][Dv];
    __shared__ float pooled[Dv];
    const int b = blockIdx.x;                              // one block per graph
    const int d = threadIdx.x & 63;
    const int g = threadIdx.x >> 6;                        // 0..3
    float acc = 0.0f;
    for (int n = g; n < Nv; n += 4)
        acc += fmaxf(Agg2[((long long)b * Nv + n) * Dv + d], 0.0f);
    part[g][d] = acc;
    __syncthreads();
    if (threadIdx.x < Dv)
        pooled[d] = (part[0][d] + part[1][d] + part[2][d] + part[3][d]) * (1.0f / (float)Nv);
    __syncthreads();
    if (threadIdx.x < 2) {
        float o = bh[threadIdx.x];
        #pragma unroll
        for (int i = 0; i < Dv; ++i) o = fmaf(pooled[i], Wh[i * 2 + threadIdx.x], o);
        Out[b * 2 + threadIdx.x] = o;
    }
}

// ---------------------------------------------------------------------------
extern "C" void kernel_launch(void* const* d_in, const int* in_sizes, int n_in,
                              void* d_out, int out_size, void* d_ws, size_t ws_size,
                              hipStream_t stream) {
    const float* x   = (const float*)d_in[0];
    const int*   ei  = (const int*)  d_in[1];   // edge_index, harness int convention
    const float* W1  = (const float*)d_in[2];
    const float* b1  = (const float*)d_in[3];
    const float* W2  = (const float*)d_in[4];
    const float* b2  = (const float*)d_in[5];
    const float* Wh  = (const float*)d_in[6];
    const float* bh  = (const float*)d_in[7];
    float* out = (float*)d_out;

    // workspace layout (bytes), all 16B aligned
    char* ws = (char*)d_ws;
    float* h0   = (float*)(ws);                              //   768 000 B
    float* dinv = (float*)(ws + 768000);                     //   256 000 B (deg -> dinv)
    float* nrm  = (float*)(ws + 1024000);                    // 4 000 000 B
    float* bufA = (float*)(ws + 5024000);                    // 16 384 000 B (t1, then t2)
    float* bufB = (float*)(ws + 21408000);                   // 16 384 000 B (agg1, then agg2)

    const int TB = 256;
    const int nNodeF   = (NUM_NODES * Fv + TB - 1) / TB;     // 750
    const int nNodes   = (NUM_NODES + TB - 1) / TB;          // 250
    const int nEdges   = (Ev + TB - 1) / TB;                 // 3907
    const int nFeat    = (NUM_NODES * Dv) / TB;              // 16000
    const int nScatter = (Ev * 16) / TB;                     // 62500
    const int nGemm2   = NUM_NODES / 128;                    // 500

    // 1. temporal mean
    k_temporal_mean<<<nNodeF, TB, 0, stream>>>(x, h0);
    // 2. degrees -> dinv
    k_deg_init<<<nNodes, TB, 0, stream>>>(dinv);
    k_deg_scatter<<<nEdges, TB, 0, stream>>>(ei, dinv);
    k_dinv<<<nNodes, TB, 0, stream>>>(dinv);
    // 3. per-edge norm
    k_norm<<<nEdges, TB, 0, stream>>>(ei, dinv, nrm);
    // 4. t1 = h0 @ W1
    k_gemm1<<<nFeat, TB, 0, stream>>>(h0, W1, bufA);
    // 5. agg1 = b1 + self + edge messages
    k_agg_init<<<nFeat, TB, 0, stream>>>(bufA, dinv, b1, bufB);
    k_edge_scatter<<<nScatter, TB, 0, stream>>>(ei, nrm, bufA, bufB);
    // 6. t2 = relu(agg1) @ W2   (WMMA f32)
    k_gemm2_wmma<<<nGemm2, TB, 0, stream>>>(bufB, W2, bufA);
    // 7. agg2 = b2 + self + edge messages
    k_agg_init<<<nFeat, TB, 0, stream>>>(bufA, dinv, b2, bufB);
    k_edge_scatter<<<nScatter, TB, 0, stream>>>(ei, nrm, bufA, bufB);
    // 8. pool + head
    k_pool_head<<<Bv, TB, 0, stream>>>(bufB, Wh, bh, out);
    (void)in_sizes; (void)n_in; (void)out_size; (void)ws_size;
}